// SpikeFeatureClassifier_31525059952998
// MI455X (gfx1250) — compile-verified
//
#include <hip/hip_runtime.h>

#define BATCH   4
#define T_DIM   256
#define N_PIX   4096
#define START_T 63
#define P_LEN   193
#define HID     16
#define FEAT    8
#define THRESH  0.9f
#define CC_ITERS 16
#define NWORDS  (N_PIX / 32)   // 128 u32 words per adjacency row

typedef __attribute__((ext_vector_type(2))) float v2f;
typedef __attribute__((ext_vector_type(8))) float v8f;

// ---------------------------------------------------------------------------
// Kernel 1: per-pixel MLP + SiLU + L2-normalize -> fn [B*N, 8] (f32, AoS)
// ---------------------------------------------------------------------------
__global__ void feat_kernel(const float* __restrict__ spike,
                            const float* __restrict__ w1,
                            const float* __restrict__ b1,
                            const float* __restrict__ w2,
                            const float* __restrict__ b2,
                            float* __restrict__ fn) {
    __shared__ float sw1[P_LEN * HID];
    __shared__ float sb1[HID];
    __shared__ float sw2[HID * FEAT];
    __shared__ float sb2[FEAT];
    int tid = threadIdx.x;
    for (int i = tid; i < P_LEN * HID; i += blockDim.x) sw1[i] = w1[i];
    for (int i = tid; i < HID * FEAT;  i += blockDim.x) sw2[i] = w2[i];
    if (tid < HID)  sb1[tid] = b1[tid];
    if (tid < FEAT) sb2[tid] = b2[tid];
    __syncthreads();

    int gid = blockIdx.x * blockDim.x + tid;    // 0 .. B*N-1
    int b   = gid >> 12;
    int pix = gid & (N_PIX - 1);

    float h[HID];
#pragma unroll
    for (int k = 0; k < HID; ++k) h[k] = sb1[k];

    const float* sp = spike + (size_t)b * T_DIM * N_PIX + (size_t)START_T * N_PIX + pix;
    for (int p = 0; p < P_LEN; ++p) {
        float x = sp[(size_t)p * N_PIX];        // coalesced across lanes
#pragma unroll
        for (int k = 0; k < HID; ++k) h[k] = fmaf(x, sw1[p * HID + k], h[k]);
    }
#pragma unroll
    for (int k = 0; k < HID; ++k) { float v = h[k]; h[k] = v / (1.f + expf(-v)); }

    float f[FEAT];
#pragma unroll
    for (int d = 0; d < FEAT; ++d) f[d] = sb2[d];
#pragma unroll
    for (int k = 0; k < HID; ++k)
#pragma unroll
        for (int d = 0; d < FEAT; ++d) f[d] = fmaf(h[k], sw2[k * FEAT + d], f[d]);

    float ss = 0.f;
#pragma unroll
    for (int d = 0; d < FEAT; ++d) { float v = f[d]; v = v / (1.f + expf(-v)); f[d] = v; ss += v * v; }
    float inv = 1.f / fmaxf(sqrtf(ss), 1e-6f);

    float* o = fn + (size_t)gid * FEAT;
#pragma unroll
    for (int d = 0; d < FEAT; ++d) o[d] = f[d] * inv;
}

// ---------------------------------------------------------------------------
// Kernel 2: WMMA f32 16x16x4 similarity tiles -> thresholded bit-packed adj.
// One wave per (16-row x 32-col) region: 4 wmma, 16 ballots, 16 word stores.
// ---------------------------------------------------------------------------
__global__ void adj_kernel(const float* __restrict__ fn, unsigned* __restrict__ adj) {
    int lane = threadIdx.x;         // 0..31, wave32, EXEC all-1s
    int li   = lane & 15;
    int half = lane >> 4;           // selects K pair {0,1} vs {2,3}
    int word  = blockIdx.x;         // column word: cols word*32 .. +31
    int rtile = blockIdx.y;         // rows rtile*16 .. +15
    int b     = blockIdx.z;

    const float* fb = fn + (size_t)b * N_PIX * FEAT;
    int rowBase = rtile * 16;
    int colBase = word * 32;

    // A (16x4): lane<16 -> K=0,1 ; lane>=16 -> K=2,3 ; M = lane&15
    v2f a0  = *(const v2f*)(fb + (size_t)(rowBase + li) * FEAT + half * 2);
    v2f a1  = *(const v2f*)(fb + (size_t)(rowBase + li) * FEAT + half * 2 + 4);
    // B (4x16) x2 column tiles, same per-lane pattern with N = lane&15
    v2f b00 = *(const v2f*)(fb + (size_t)(colBase      + li) * FEAT + half * 2);
    v2f b01 = *(const v2f*)(fb + (size_t)(colBase      + li) * FEAT + half * 2 + 4);
    v2f b10 = *(const v2f*)(fb + (size_t)(colBase + 16 + li) * FEAT + half * 2);
    v2f b11 = *(const v2f*)(fb + (size_t)(colBase + 16 + li) * FEAT + half * 2 + 4);

    v8f c0 = {};
    c0 = __builtin_amdgcn_wmma_f32_16x16x4_f32(false, a0, false, b00, (short)0, c0, false, false);
    c0 = __builtin_amdgcn_wmma_f32_16x16x4_f32(false, a1, false, b01, (short)0, c0, false, false);
    v8f c1 = {};
    c1 = __builtin_amdgcn_wmma_f32_16x16x4_f32(false, a0, false, b10, (short)0, c1, false, false);
    c1 = __builtin_amdgcn_wmma_f32_16x16x4_f32(false, a1, false, b11, (short)0, c1, false, false);

    // C layout: VGPR v, lanes 0-15 -> (row v, col lane); lanes 16-31 -> (row v+8, col lane-16)
    unsigned myword = 0;
#pragma unroll
    for (int v = 0; v < 8; ++v) {
        unsigned m0 = __builtin_amdgcn_ballot_w32(c0[v] >= THRESH);
        unsigned m1 = __builtin_amdgcn_ballot_w32(c1[v] >= THRESH);
        unsigned lowW  = (m0 & 0xffffu) | (m1 << 16);         // row rowBase+v
        unsigned highW = (m0 >> 16) | (m1 & 0xffff0000u);     // row rowBase+v+8
        if ((lane & 7) == v) myword = (lane & 8) ? highW : lowW;
    }
    if (lane < 16) {
        int row = rowBase + (lane & 7) + ((lane & 8) ? 8 : 0);
        adj[(size_t)b * N_PIX * NWORDS + (size_t)row * NWORDS + word] = myword;
    }
}

// ---------------------------------------------------------------------------
// Connected components: min-label over adjacency + pointer jumping
// ---------------------------------------------------------------------------
__global__ void init_labels(int* __restrict__ labels) {
    int gid = blockIdx.x * blockDim.x + threadIdx.x;
    labels[gid] = gid & (N_PIX - 1);
}

__global__ void cc_min(const unsigned* __restrict__ adj, const int* __restrict__ labels,
                       int* __restrict__ newlab) {
    __shared__ int slab[N_PIX];
    int b   = blockIdx.y;
    int tid = threadIdx.x;
    const int* lb = labels + b * N_PIX;
    for (int i = tid; i < N_PIX; i += blockDim.x) slab[i] = lb[i];
    __syncthreads();

    int row = blockIdx.x * blockDim.x + tid;
    const unsigned* arow = adj + (size_t)b * N_PIX * NWORDS + (size_t)row * NWORDS;
    int m = N_PIX;                              // reference fill value
    for (int wd = 0; wd < NWORDS; ++wd) {
        unsigned w = arow[wd];
        int base = wd * 32;
        while (w) {
            int j = base + __builtin_ctz(w);
            w &= w - 1;
            int l = slab[j];
            m = l < m ? l : m;
        }
    }
    newlab[b * N_PIX + row] = m;
}

__global__ void cc_jump(const int* __restrict__ newlab, int* __restrict__ labels) {
    int gid = blockIdx.x * blockDim.x + threadIdx.x;
    int b = gid >> 12;
    const int* nb = newlab + b * N_PIX;
    int l = nb[gid & (N_PIX - 1)];
    l = l < N_PIX ? l : N_PIX - 1;              // jax-style clamp
    labels[gid] = nb[l];
}

// ---------------------------------------------------------------------------
// Dense component ids: rank = cumsum(is_root)-1 ; comp = rank[label]
// One 1024-thread block per batch.
// ---------------------------------------------------------------------------
__global__ void comp_kernel(const int* __restrict__ labels, int* __restrict__ comp) {
    __shared__ int part[1024];
    __shared__ int rankL[N_PIX];
    int b = blockIdx.x, t = threadIdx.x;
    const int* lb = labels + b * N_PIX;
    int base = t * 4;
    int l0 = lb[base], l1 = lb[base + 1], l2 = lb[base + 2], l3 = lb[base + 3];
    int r0 = (l0 == base), r1 = (l1 == base + 1), r2 = (l2 == base + 2), r3 = (l3 == base + 3);
    int s = r0 + r1 + r2 + r3;
    part[t] = s;
    __syncthreads();
    for (int off = 1; off < 1024; off <<= 1) {
        int v = part[t];
        int add = (t >= off) ? part[t - off] : 0;
        __syncthreads();
        part[t] = v + add;
        __syncthreads();
    }
    int run = part[t] - s;                      // exclusive prefix
    run += r0; rankL[base]     = run - 1;
    run += r1; rankL[base + 1] = run - 1;
    run += r2; rankL[base + 2] = run - 1;
    run += r3; rankL[base + 3] = run - 1;
    __syncthreads();
    int* cb = comp + b * N_PIX;
    cb[base]     = rankL[l0];
    cb[base + 1] = rankL[l1];
    cb[base + 2] = rankL[l2];
    cb[base + 3] = rankL[l3];
}

// ---------------------------------------------------------------------------
// Output: zero-fill 268 MB then scatter ones at [b, comp[i], i]
// ---------------------------------------------------------------------------
__global__ void zero_kernel(float4* __restrict__ out, long n4) {
    long gid = blockIdx.x * (long)blockDim.x + threadIdx.x;
    long stride = (long)gridDim.x * blockDim.x;
    float4 z; z.x = 0.f; z.y = 0.f; z.z = 0.f; z.w = 0.f;
    for (long i = gid; i < n4; i += stride) out[i] = z;
}

__global__ void scatter_kernel(const int* __restrict__ comp, float* __restrict__ out) {
    int gid = blockIdx.x * blockDim.x + threadIdx.x;   // B*N
    int b = gid >> 12, i = gid & (N_PIX - 1);
    int c = comp[gid];
    out[(size_t)b * N_PIX * N_PIX + (size_t)c * N_PIX + i] = 1.0f;
}

// ---------------------------------------------------------------------------
extern "C" void kernel_launch(void* const* d_in, const int* in_sizes, int n_in,
                              void* d_out, int out_size, void* d_ws, size_t ws_size,
                              hipStream_t stream) {
    const float* spike = (const float*)d_in[0];
    const float* w1    = (const float*)d_in[1];
    const float* b1    = (const float*)d_in[2];
    const float* w2    = (const float*)d_in[3];
    const float* b2    = (const float*)d_in[4];
    float* out = (float*)d_out;

    // workspace layout (~9.1 MB total)
    char* ws = (char*)d_ws;
    float*    fn     = (float*)ws;                                  // 512 KB
    unsigned* adj    = (unsigned*)(ws + 524288);                    // 8 MB
    int*      labels = (int*)(ws + 524288 + 8388608);               // 64 KB
    int*      newlab = labels + BATCH * N_PIX;                      // 64 KB
    int*      comp   = newlab + BATCH * N_PIX;                      // 64 KB

    zero_kernel<<<2048, 256, 0, stream>>>((float4*)out, (long)BATCH * N_PIX * N_PIX / 4);
    feat_kernel<<<(BATCH * N_PIX) / 256, 256, 0, stream>>>(spike, w1, b1, w2, b2, fn);

    dim3 g(NWORDS, N_PIX / 16, BATCH);        // 128 x 256 x 4 waves
    adj_kernel<<<g, 32, 0, stream>>>(fn, adj);

    init_labels<<<(BATCH * N_PIX) / 256, 256, 0, stream>>>(labels);
    for (int it = 0; it < CC_ITERS; ++it) {
        cc_min<<<dim3(N_PIX / 256, BATCH), 256, 0, stream>>>(adj, labels, newlab);
        cc_jump<<<(BATCH * N_PIX) / 256, 256, 0, stream>>>(newlab, labels);
    }
    comp_kernel<<<BATCH, 1024, 0, stream>>>(labels, comp);
    scatter_kernel<<<(BATCH * N_PIX) / 256, 256, 0, stream>>>(comp, out);
}